// VectorQuantizerEMA1_26972394619049
// MI455X (gfx1250) — compile-verified
//
#include <hip/hip_runtime.h>
#include <math.h>

// ---------------------------------------------------------------------------
// VQ-VAE vector quantizer (eval forward) for MI455X / gfx1250.
//   inputs : [32,32,32,64] f32  -> N=32768 rows, D=64
//   weight : [1024,64]     f32  -> K=1024 codes
// Outputs concatenated in d_out (f32):
//   quantized_st [N*64] | encodings [N*1024] | commitment_loss | perplexity
// Matmul via V_WMMA_F32_16X16X4_F32 (full fp32 precision; compute is only
// ~4.3 GFLOP, the kernel is bound by the 128 MB one-hot store traffic).
// ---------------------------------------------------------------------------

#define DIM        64
#define KCODES     1024
#define NROWS      32768
#define NGROUPS    (NROWS / 16)          // 2048 row-groups, one per wave
#define WPB        4                     // waves per block

// ws layout (floats)
#define WS_ENORM   0
#define WS_COUNTS  1024
#define WS_PART    2048                  // 2048 per-wave loss partials

typedef float v2f __attribute__((ext_vector_type(2)));
typedef float v8f __attribute__((ext_vector_type(8)));

__global__ void vq_init(float* ws) {
    int t = blockIdx.x * blockDim.x + threadIdx.x;
    if (t < KCODES) ws[WS_COUNTS + t] = 0.0f;
}

__global__ void vq_enorm(const float* __restrict__ w, float* ws) {
    int k = blockIdx.x * blockDim.x + threadIdx.x;
    if (k < KCODES) {
        const float* r = w + k * DIM;
        float s = 0.0f;
        #pragma unroll
        for (int d = 0; d < DIM; ++d) { float v = r[d]; s += v * v; }
        ws[WS_ENORM + k] = s;
    }
}

__global__ __launch_bounds__(WPB * 32)
void vq_main(const float* __restrict__ x, const float* __restrict__ w,
             float* __restrict__ outQuant, float* __restrict__ outEnc,
             float* __restrict__ ws)
{
    const int lane = threadIdx.x & 31;
    const int wave = threadIdx.x >> 5;
    const int grp  = blockIdx.x * WPB + wave;     // 0..2047
    const int h    = lane >> 4;                   // K-half select
    const int l15  = lane & 15;

    // ---- preload A fragments: row (grp*16 + l15), K-pairs 4t + 2h + {0,1} ----
    const float* xrow = x + (size_t)(grp * 16 + l15) * DIM + 2 * h;
    v2f xa[16];
    float xpart = 0.0f;
    #pragma unroll
    for (int t = 0; t < 16; ++t) {
        float2 v = *(const float2*)(xrow + 4 * t);
        xa[t].x = v.x; xa[t].y = v.y;
        xpart += v.x * v.x + v.y * v.y;
    }
    // full ||x_row||^2: combine the two K-halves (lanes L and L^16 share a row)
    float xnorm = xpart + __shfl_xor(xpart, 16);  // lane holds norm of row l15
    float xn[8];
    #pragma unroll
    for (int v = 0; v < 8; ++v) xn[v] = __shfl(xnorm, v + 8 * h); // norm of C-row m=v+8h

    float bestVal[8]; int bestIdx[8];
    #pragma unroll
    for (int v = 0; v < 8; ++v) { bestVal[v] = 3.4e38f; bestIdx[v] = 0; }

    const float* eN = ws + WS_ENORM;

    for (int tile = 0; tile < KCODES / 16; ++tile) {
        const int n = tile * 16 + l15;            // codebook row for this lane's column
        const float* wrow = w + (size_t)n * DIM + 2 * h;
        v8f c = {};
        #pragma unroll
        for (int t = 0; t < 16; ++t) {
            float2 bv = *(const float2*)(wrow + 4 * t);
            v2f b; b.x = bv.x; b.y = bv.y;
            c = __builtin_amdgcn_wmma_f32_16x16x4_f32(
                    /*neg_a=*/false, xa[t], /*neg_b=*/false, b,
                    /*c_mod=*/(short)0, c, /*reuse_a=*/false, /*reuse_b=*/false);
        }
        const float en = eN[n];
        #pragma unroll
        for (int v = 0; v < 8; ++v) {
            float dv = xn[v] + en - 2.0f * c[v];
            if (dv < bestVal[v]) { bestVal[v] = dv; bestIdx[v] = n; } // strict < keeps first idx
        }
    }

    // ---- argmin across the 16 columns held by the lanes of each half-wave ----
    #pragma unroll
    for (int mask = 1; mask < 16; mask <<= 1) {
        #pragma unroll
        for (int v = 0; v < 8; ++v) {
            float ov = __shfl_xor(bestVal[v], mask);
            int   oi = __shfl_xor(bestIdx[v], mask);
            if (ov < bestVal[v] || (ov == bestVal[v] && oi < bestIdx[v])) {
                bestVal[v] = ov; bestIdx[v] = oi;
            }
        }
    }
    // broadcast the 16 row winners to every lane (row r lives in half r>>3, slot r&7)
    int rowIdx[16];
    #pragma unroll
    for (int r = 0; r < 16; ++r)
        rowIdx[r] = __shfl(bestIdx[r & 7], (r >> 3) * 16);

    // ---- quantized rows + commitment-loss partial (float2 per lane per row) ----
    float lsum = 0.0f;
    #pragma unroll
    for (int r = 0; r < 16; ++r) {
        const int R   = grp * 16 + r;
        const int idx = rowIdx[r];
        float2 wv = *(const float2*)(w + (size_t)idx * DIM + lane * 2);
        float2 xv = *(const float2*)(x + (size_t)R   * DIM + lane * 2);
        *(float2*)(outQuant + (size_t)R * DIM + lane * 2) = wv;
        float dx = wv.x - xv.x, dy = wv.y - xv.y;
        lsum += dx * dx + dy * dy;
    }

    // ---- one-hot encodings: 32 columns per lane per row, b128 stores ----
    #pragma unroll
    for (int r = 0; r < 16; ++r) {
        const int R   = grp * 16 + r;
        const int idx = rowIdx[r];
        float* base = outEnc + (size_t)R * KCODES + lane * 32;
        #pragma unroll
        for (int j = 0; j < 8; ++j) {
            const int cc = lane * 32 + j * 4;
            float4 o;
            o.x = (cc + 0 == idx) ? 1.0f : 0.0f;
            o.y = (cc + 1 == idx) ? 1.0f : 0.0f;
            o.z = (cc + 2 == idx) ? 1.0f : 0.0f;
            o.w = (cc + 3 == idx) ? 1.0f : 0.0f;
            *(float4*)(base + j * 4) = o;
        }
    }

    // ---- wave reduction of loss, deterministic partial store; count atomics ----
    #pragma unroll
    for (int mask = 1; mask < 32; mask <<= 1) lsum += __shfl_xor(lsum, mask);
    if (lane == 0) {
        ws[WS_PART + grp] = lsum;
        float* counts = ws + WS_COUNTS;
        for (int r = 0; r < 16; ++r)
            atomicAdd(&counts[rowIdx[r]], 1.0f);  // integer-valued f32 adds: exact, order-free
    }
}

__global__ __launch_bounds__(1024)
void vq_final(const float* __restrict__ ws, float* __restrict__ outScal) {
    __shared__ float sE[1024];
    __shared__ float sL[1024];
    const int t = threadIdx.x;
    float p = ws[WS_COUNTS + t] * (1.0f / (float)NROWS);
    sE[t] = p * logf(p + 1e-10f);
    sL[t] = ws[WS_PART + t] + ws[WS_PART + 1024 + t];
    __syncthreads();
    for (int s = 512; s > 0; s >>= 1) {
        if (t < s) { sE[t] += sE[t + s]; sL[t] += sL[t + s]; }
        __syncthreads();
    }
    if (t == 0) {
        outScal[0] = sL[0] * (1.0f / ((float)NROWS * (float)DIM)); // commitment loss
        outScal[1] = expf(-sE[0]);                                 // perplexity
    }
}

extern "C" void kernel_launch(void* const* d_in, const int* in_sizes, int n_in,
                              void* d_out, int out_size, void* d_ws, size_t ws_size,
                              hipStream_t stream) {
    const float* x = (const float*)d_in[0];   // [32,32,32,64] f32
    const float* w = (const float*)d_in[1];   // [1024,64] f32
    float* out = (float*)d_out;
    float* ws  = (float*)d_ws;

    float* outQuant = out;                                   // 2,097,152 floats
    float* outEnc   = out + (size_t)NROWS * DIM;             // 33,554,432 floats
    float* outScal  = outEnc + (size_t)NROWS * KCODES;       // loss, perplexity

    vq_init <<<4, 256, 0, stream>>>(ws);
    vq_enorm<<<4, 256, 0, stream>>>(w, ws);
    vq_main <<<NGROUPS / WPB, WPB * 32, 0, stream>>>(x, w, outQuant, outEnc, ws);
    vq_final<<<1, 1024, 0, stream>>>(ws, outScal);
}